// VideoHungarianMatcherProjPair_69535520522305
// MI455X (gfx1250) — compile-verified
//
#include <hip/hip_runtime.h>
#include <hip/hip_bf16.h>
#include <math.h>

typedef __attribute__((ext_vector_type(2))) float v2f;
typedef __attribute__((ext_vector_type(8))) float v8f;

#define Qn 100
#define Gn 16
#define Tn 2
#define Hn 128
#define Wn 128
#define Cln 81
#define QP 112   // Q padded to 7 tiles of 16
#define NQT 7
#define NB 8
#define THR 0.3f

#if defined(__gfx1250__)
#if __has_builtin(__builtin_amdgcn_wmma_f32_16x16x4_f32)
#define HAVE_WMMA_F32X4 1
#endif
#endif

__device__ __forceinline__ float logsig(float v) {
    // log(sigmoid(v)) = min(v,0) - log(1 + exp(-|v|)), stable
    return fminf(v, 0.f) - __logf(1.f + __expf(-fabsf(v)));
}

// D = A(16x4 f32) * B(4x16 f32) + C(16x16 f32), wave32 WMMA
__device__ __forceinline__ v8f wmma_f32_4(v2f a, v2f b, v8f c) {
#ifdef HAVE_WMMA_F32X4
    return __builtin_amdgcn_wmma_f32_16x16x4_f32(
        /*neg_a=*/false, a, /*neg_b=*/false, b,
        /*c_mod=*/(short)0, c, /*reuse_a=*/false, /*reuse_b=*/false);
#else
    // layout-equivalent shuffle fallback (compile safety net only)
    int lane = threadIdx.x & 31;
    int N = lane & 15;
    #pragma unroll
    for (int v = 0; v < 8; ++v) {
        int M = v + ((lane >> 4) << 3);
        float sum = c[v];
        #pragma unroll
        for (int k = 0; k < 4; ++k) {
            float av = __shfl(a[k & 1], ((k >> 1) << 4) + M, 32);
            float bv = __shfl(b[k & 1], ((k >> 1) << 4) + N, 32);
            sum = fmaf(av, bv, sum);
        }
        c[v] = sum;
    }
    return c;
#endif
}

// ---------------------------------------------------------------------------
// Pairwise numerator: one block per (t, h). Fused src_sim -> WMMA, no
// materialization of the (Q,T,8,H,W) tensor. logsig precomputed in LDS
// (reused ~8x per staged value); inner loop is branchless so EXEC stays
// all-1s around every WMMA. Partial numerators per (t,h,q,g) go to
// workspace for a deterministic fixed-order reduction.
// ---------------------------------------------------------------------------
__launch_bounds__(256)
__global__ void pairwise_kernel(const float* __restrict__ out_mask,
                                const float* __restrict__ tgt_sim,
                                const int*   __restrict__ box,
                                float* __restrict__ partialNum) {
    const int bh = blockIdx.x;           // [0, Tn*Hn)
    const int t  = bh / Hn;
    const int h  = bh % Hn;
    const int tid  = threadIdx.x;
    const int lane = tid & 31;
    const int wv   = tid >> 5;           // wave id == neighbor id

    __shared__ float         xRows[16][3][Wn];   // 24 KB: out_mask rows (q-tile)
    __shared__ float         lfRows[16][3][Wn];  // 24 KB: logsig of staged rows
    __shared__ unsigned char tB[NB][Wn][Gn];     // 16 KB: thresholded tgt bits
    __shared__ float         ldsAcc[NB][256];    //  8 KB: per-wave 16x16 partials

    // Stage tgt bits once per block: tB[nb][w][g]
    for (int idx = tid; idx < Gn * NB * Wn; idx += 256) {
        int w  = idx & (Wn - 1);
        int gn = idx >> 7;               // g*8 + nb
        int g = gn >> 3, nb = gn & 7;
        float sim = tgt_sim[(((g * Tn + t) * NB + nb) * Hn + h) * Wn + w];
        int   bx  = box[((g * Tn + t) * Hn + h) * Wn + w];
        tB[nb][w][g] = (sim >= THR && bx != 0) ? 1 : 0;
    }

    // Row-major 3x3 (dil=2) neighbor offsets, center removed
    const int diTab[8] = {-2, -2, -2,  0, 0,  2, 2, 2};
    const int djTab[8] = {-2,  0,  2, -2, 2, -2, 0, 2};
    const int di = diTab[wv], dj = djTab[wv];
    const int r  = di / 2 + 1;                     // staged row index 0..2
    const bool rowOK = (h + di >= 0) && (h + di < Hn);  // wave-uniform

    const int m     = lane & 15;                   // A: M row; B: N (=g) col
    const int khalf = (lane >> 4) << 1;            // 0 or 2 (K half-select)

    for (int qt = 0; qt < NQT; ++qt) {
        __syncthreads();
        // Stage out_mask rows h-2, h, h+2 for this q-tile (zero-padded),
        // and precompute logsig once per staged value.
        for (int idx = tid; idx < 16 * 3 * Wn; idx += 256) {
            int w  = idx & (Wn - 1);
            int qr = idx >> 7;                     // qi*3 + rr, [0,48)
            int qi = qr / 3, rr = qr % 3;
            int q  = qt * 16 + qi;
            int hh = h + (rr - 1) * 2;
            float v = 0.f;
            if (q < Qn && hh >= 0 && hh < Hn)
                v = out_mask[((q * Tn + t) * Hn + hh) * Wn + w];
            xRows[qi][rr][w]  = v;
            lfRows[qi][rr][w] = logsig(v);
        }
        __syncthreads();

        v8f acc = {};
        #pragma unroll 2
        for (int base = 0; base < Wn; base += 4) {
            v2f a, b;
            #pragma unroll
            for (int u = 0; u < 2; ++u) {
                int w   = base + khalf + u;        // this lane's K (= pixel w)
                int wn  = w + dj;
                int wnc = min(max(wn, 0), Wn - 1); // clamped, branchless
                float xc  = xRows[m][1][w];
                float xn  = xRows[m][r][wnc];
                float lfc = lfRows[m][1][w];
                float lfn = lfRows[m][r][wnc];
                float s  = xc + xn;                // lb = lf - x
                float a1 = lfc + lfn;
                float av = -(fmaxf(a1, a1 - s) + __logf(1.f + __expf(-fabsf(s))));
                bool inb = rowOK && (wn >= 0) && (wn < Wn);
                a[u] = inb ? av : 0.f;             // v_cndmask, no branch
                b[u] = (float)tB[wv][w][m];
            }
            acc = wmma_f32_4(a, b, acc);
        }

        // Store per C/D layout: VGPR v -> M = v + 8*(lane>=16), N = lane%16
        #pragma unroll
        for (int v = 0; v < 8; ++v)
            ldsAcc[wv][(v + ((lane >> 4) << 3)) * 16 + (lane & 15)] = acc[v];
        __syncthreads();

        // Fixed-order cross-wave (= cross-neighbor) reduction, unique write
        {
            int e = tid;                           // exactly 256 elements
            float sum = 0.f;
            #pragma unroll
            for (int wvi = 0; wvi < NB; ++wvi) sum += ldsAcc[wvi][e];
            int qq = qt * 16 + (e >> 4);
            int gg = e & 15;
            partialNum[((t * Hn + h) * QP + qq) * Gn + gg] = sum;
        }
    }
}

// ---------------------------------------------------------------------------
// Denominator: clamp(sum of tgt bits over c, 1, inf) per (g, t)
// ---------------------------------------------------------------------------
__global__ void denom_kernel(const float* __restrict__ tgt_sim,
                             const int*   __restrict__ box,
                             float* __restrict__ den) {
    int b = blockIdx.x;                  // [0, Gn*Tn)
    int g = b >> 1, t = b & 1;
    float s = 0.f;
    for (int idx = threadIdx.x; idx < NB * Hn * Wn; idx += 256) {
        int nb = idx >> 14;
        int hw = idx & 16383;
        float sim = tgt_sim[((g * Tn + t) * NB + nb) * Hn * Wn + hw];
        int   bx  = box[(g * Tn + t) * Hn * Wn + hw];
        s += (sim >= THR && bx != 0) ? 1.f : 0.f;
    }
    __shared__ float red[256];
    red[threadIdx.x] = s;
    __syncthreads();
    for (int o = 128; o > 0; o >>= 1) {
        if (threadIdx.x < o) red[threadIdx.x] += red[threadIdx.x + o];
        __syncthreads();
    }
    if (threadIdx.x == 0) den[g * Tn + t] = fmaxf(red[0], 1.f);
}

// ---------------------------------------------------------------------------
// Axis projections of sigmoid(out_mask): sigmoid is monotone -> sigmoid(max)
// ---------------------------------------------------------------------------
__global__ void proj_src_kernel(const float* __restrict__ out_mask,
                                float* __restrict__ projW,   // (Q, T*H)
                                float* __restrict__ projH) { // (Q, T*W)
    int q = blockIdx.x, tid = threadIdx.x;   // 256 = Tn * 128
    int t = tid >> 7, x = tid & 127;
    const float* p = out_mask + (size_t)(q * Tn + t) * Hn * Wn;
    float mw = -INFINITY;
    for (int w = 0; w < Wn; ++w) mw = fmaxf(mw, p[x * Wn + w]);
    projW[q * 256 + t * Hn + x] = 1.f / (1.f + __expf(-mw));
    float mh = -INFINITY;
    for (int hh = 0; hh < Hn; ++hh) mh = fmaxf(mh, p[hh * Wn + x]);
    projH[q * 256 + t * Wn + x] = 1.f / (1.f + __expf(-mh));
}

__global__ void proj_tgt_kernel(const int* __restrict__ box,
                                float* __restrict__ projW,   // (G, T*H)
                                float* __restrict__ projH) { // (G, T*W)
    int g = blockIdx.x, tid = threadIdx.x;
    int t = tid >> 7, x = tid & 127;
    const int* p = box + (size_t)(g * Tn + t) * Hn * Wn;
    int mw = 0;
    for (int w = 0; w < Wn; ++w) mw = max(mw, p[x * Wn + w]);
    projW[g * 256 + t * Hn + x] = (float)mw;
    int mh = 0;
    for (int hh = 0; hh < Hn; ++hh) mh = max(mh, p[hh * Wn + x]);
    projH[g * 256 + t * Wn + x] = (float)mh;
}

// ---------------------------------------------------------------------------
// cost_class = -softmax(pred_logits)[q, tgt_ids[g]]
// ---------------------------------------------------------------------------
__global__ void class_kernel(const float* __restrict__ logits,
                             const int* __restrict__ ids,
                             float* __restrict__ cc) {
    int q = blockIdx.x, tid = threadIdx.x;   // 128 threads
    __shared__ float red[128];
    float mx = -INFINITY;
    for (int c = tid; c < Cln; c += 128) mx = fmaxf(mx, logits[q * Cln + c]);
    red[tid] = mx;
    __syncthreads();
    for (int o = 64; o > 0; o >>= 1) {
        if (tid < o) red[tid] = fmaxf(red[tid], red[tid + o]);
        __syncthreads();
    }
    mx = red[0];
    __syncthreads();
    float s = 0.f;
    for (int c = tid; c < Cln; c += 128) s += __expf(logits[q * Cln + c] - mx);
    red[tid] = s;
    __syncthreads();
    for (int o = 64; o > 0; o >>= 1) {
        if (tid < o) red[tid] += red[tid + o];
        __syncthreads();
    }
    s = red[0];
    if (tid < Gn) {
        int c = ids[tid];
        cc[q * Gn + tid] = -__expf(logits[q * Cln + c] - mx) / s;
    }
}

// ---------------------------------------------------------------------------
// Combine: dice(projW) + dice(projH) + mean_t(num/den) + class, fixed order
// ---------------------------------------------------------------------------
__global__ void combine_kernel(const float* __restrict__ projSrcW,
                               const float* __restrict__ projSrcH,
                               const float* __restrict__ projTgtW,
                               const float* __restrict__ projTgtH,
                               const float* __restrict__ partialNum,
                               const float* __restrict__ den,
                               const float* __restrict__ cc,
                               float* __restrict__ out) {
    int q = blockIdx.x, tid = threadIdx.x;   // 64 threads
    __shared__ float sW[256], sH[256];
    for (int i = tid; i < 256; i += 64) {
        sW[i] = projSrcW[q * 256 + i];
        sH[i] = projSrcH[q * 256 + i];
    }
    __syncthreads();
    if (tid < Gn) {
        int g = tid;
        float numW = 0.f, ssW = 0.f, stW = 0.f;
        float numH = 0.f, ssH = 0.f, stH = 0.f;
        for (int c = 0; c < 256; ++c) {
            float s1 = sW[c], t1 = projTgtW[g * 256 + c];
            numW += s1 * t1; ssW += s1; stW += t1;
            float s2 = sH[c], t2 = projTgtH[g * 256 + c];
            numH += s2 * t2; ssH += s2; stH += t2;
        }
        float diceW = 1.f - (2.f * numW + 1.f) / (ssW + stW + 1.f);
        float diceH = 1.f - (2.f * numH + 1.f) / (ssH + stH + 1.f);
        float pair = 0.f;
        for (int t = 0; t < Tn; ++t) {
            float num = 0.f;
            for (int h = 0; h < Hn; ++h)
                num += partialNum[((t * Hn + h) * QP + q) * Gn + g];
            pair += num / den[g * Tn + t];
        }
        pair *= 0.5f;   // mean over T=2; WARMUP = 1
        out[q * Gn + g] = cc[q * Gn + g] + diceW + diceH + pair;
    }
}

extern "C" void kernel_launch(void* const* d_in, const int* in_sizes, int n_in,
                              void* d_out, int out_size, void* d_ws, size_t ws_size,
                              hipStream_t stream) {
    const float* pred_logits = (const float*)d_in[0];   // (Q, 81)
    const float* out_mask    = (const float*)d_in[1];   // (Q, T, H, W)
    const float* tgt_sim     = (const float*)d_in[2];   // (G, T, 8, H, W)
    const int*   tgt_ids     = (const int*)  d_in[3];   // (G,)
    const int*   box         = (const int*)  d_in[4];   // (G, T, H, W)
    float* out = (float*)d_out;                          // (Q, G)

    float* ws = (float*)d_ws;
    float* partialNum = ws;                 ws += (size_t)Tn * Hn * QP * Gn;  // 458752
    float* den        = ws;                 ws += Gn * Tn;                    // 32
    float* projSrcW   = ws;                 ws += Qn * 256;                   // 25600
    float* projSrcH   = ws;                 ws += Qn * 256;
    float* projTgtW   = ws;                 ws += Gn * 256;                   // 4096
    float* projTgtH   = ws;                 ws += Gn * 256;
    float* cc         = ws;                 ws += Qn * Gn;                    // 1600

    class_kernel   <<<Qn,       128, 0, stream>>>(pred_logits, tgt_ids, cc);
    proj_src_kernel<<<Qn,       256, 0, stream>>>(out_mask, projSrcW, projSrcH);
    proj_tgt_kernel<<<Gn,       256, 0, stream>>>(box, projTgtW, projTgtH);
    denom_kernel   <<<Gn * Tn,  256, 0, stream>>>(tgt_sim, box, den);
    pairwise_kernel<<<Tn * Hn,  256, 0, stream>>>(out_mask, tgt_sim, box, partialNum);
    combine_kernel <<<Qn,        64, 0, stream>>>(projSrcW, projSrcH, projTgtW, projTgtH,
                                                  partialNum, den, cc, out);
}